// Nms_11132555231477
// MI455X (gfx1250) — compile-verified
//
#include <hip/hip_runtime.h>
#include <hip/hip_bf16.h>

#define NMS_N   4096
#define NWORDS  (NMS_N / 32)       // 128 words (512 B) per row of the suppression bitmask
#define THR     0.5f
#define SWEEP_D 8                  // async-to-LDS ring depth (rows in flight)

// ---------------------------------------------------------------------------
// Kernel 1: bitonic sort of (score, idx) keys in LDS.
// Key = (~float_bits(score) << 32) | idx  -> ascending u64 sort == descending
// score with stable (ascending idx) tie-break, matching jnp.argsort(-scores).
// Scores are in [0,1) so the sign bit is 0 and uint compare is monotonic.
// 4096 keys * 8B = 32KB LDS (WGP has 320KB). 1024 threads, 4 elems/thread.
// ---------------------------------------------------------------------------
__global__ __launch_bounds__(1024) void nms_sort(const float* __restrict__ scores,
                                                 const float4* __restrict__ rois4,
                                                 unsigned* __restrict__ order,
                                                 float4* __restrict__ boxes)
{
    __shared__ unsigned long long keys[NMS_N];
    const unsigned tid = threadIdx.x;

    #pragma unroll
    for (int m = 0; m < 4; ++m) {
        unsigned i = tid + m * 1024u;
        unsigned bits = __float_as_uint(scores[i]);
        keys[i] = ((unsigned long long)(~bits) << 32) | (unsigned long long)i;
    }
    __syncthreads();

    for (unsigned k = 2; k <= NMS_N; k <<= 1) {
        for (unsigned j = k >> 1; j > 0; j >>= 1) {
            #pragma unroll
            for (int m = 0; m < 4; ++m) {
                unsigned i   = tid + m * 1024u;
                unsigned ixj = i ^ j;
                if (ixj > i) {                       // each disjoint pair handled once
                    bool up = ((i & k) == 0u);
                    unsigned long long a = keys[i];
                    unsigned long long b = keys[ixj];
                    if ((a > b) == up) { keys[i] = b; keys[ixj] = a; }
                }
            }
            __syncthreads();
        }
    }

    #pragma unroll
    for (int m = 0; m < 4; ++m) {
        unsigned i = tid + m * 1024u;
        unsigned o = (unsigned)(keys[i] & 0xffffffffu);
        order[i] = o;
        boxes[i] = rois4[o];                          // gather boxes into sorted order
    }
}

// ---------------------------------------------------------------------------
// Kernel 2: build upper-triangular suppression bitmask.
// mask[i][w] bit b == 1  iff  j = 32w+b > i  and  IoU(box_i, box_j) > THR.
// IoU test done division-free: inter > THR * max(union, 1e-9).
// One thread per (i, w): 4096*128 threads; lower-triangle words early-out.
// 2MB of b32 stores; boxes stream from L2 (trivially cached re-reads).
// ---------------------------------------------------------------------------
__global__ __launch_bounds__(256) void nms_mask(const float4* __restrict__ boxes,
                                                unsigned* __restrict__ mask)
{
    unsigned gid = blockIdx.x * 256u + threadIdx.x;   // 0 .. N*NWORDS-1
    unsigned i   = gid >> 7;                          // row
    unsigned w   = gid & (NWORDS - 1);                // 32-col word
    unsigned jbase = w << 5;

    unsigned bits = 0u;
    if (jbase + 31u > i) {                            // word entirely <= i -> all zero
        float4 bi = boxes[i];
        float  ai = (bi.z - bi.x) * (bi.w - bi.y);
        #pragma unroll 8
        for (unsigned b = 0; b < 32; ++b) {
            unsigned j = jbase + b;
            float4 bj = boxes[j];
            float  aj = (bj.z - bj.x) * (bj.w - bj.y);
            float iw = fminf(bi.z, bj.z) - fmaxf(bi.x, bj.x);
            float ih = fminf(bi.w, bj.w) - fmaxf(bi.y, bj.y);
            iw = fmaxf(iw, 0.0f);
            ih = fmaxf(ih, 0.0f);
            float inter = iw * ih;
            float uni   = fmaxf(ai + aj - inter, 1e-9f);
            bool  sup   = (inter > THR * uni) && (j > i);
            bits |= (sup ? 1u : 0u) << b;
        }
    }
    mask[gid] = bits;
}

// ---------------------------------------------------------------------------
// Kernel 3: serial greedy sweep, single wave32.
// * 128-word "removed" accumulator: lane l owns the CONTIGUOUS words
//   4l..4l+3 in a uint4 of VGPRs, so each row is consumed with a single
//   bank-conflict-free ds_load_b128 at lane*16 (exactly the bytes the async
//   engine deposited).
// * Every mask row is streamed Global->LDS with the CDNA5 async-to-LDS
//   engine into an 8-slot ring: one global_load_async_to_lds_b128 per row
//   (32 lanes x 16B = 512B). Async loads complete IN ORDER (ISA 08 §4.1),
//   so with SWEEP_D rows outstanding, s_wait_asynccnt (SWEEP_D-1)
//   guarantees row i is LDS-resident before the conditional OR — the full
//   L2 latency hides behind the 8-row pipeline on this inherently serial
//   scan.
// * Removed-bit test is all-uniform (owner = wi>>2, component = wi&3)
//   -> v_readlane path, no barriers anywhere (wave32 lockstep).
// ---------------------------------------------------------------------------
__global__ __launch_bounds__(32) void nms_sweep(const unsigned* __restrict__ mask,
                                                unsigned* __restrict__ keepw)
{
    __shared__ unsigned buf[SWEEP_D][NWORDS];         // 8 x 512B ring
    const int lane = threadIdx.x;                     // 0..31
    uint4 racc = make_uint4(0u, 0u, 0u, 0u);          // removed words 4*lane..4*lane+3

    // Low 32 bits of a flat shared-aperture pointer == LDS byte offset.
    const unsigned lds0 = (unsigned)(uintptr_t)(&buf[0][0]) + (unsigned)lane * 16u;

    // Prime the pipeline: rows 0..SWEEP_D-1 in flight.
    #pragma unroll
    for (int p = 0; p < SWEEP_D; ++p) {
        unsigned ldsa = lds0 + (unsigned)p * (NWORDS * 4u);
        unsigned long long ga =
            (unsigned long long)(uintptr_t)(mask + (size_t)p * NWORDS) +
            (unsigned long long)(lane * 16);
        asm volatile("global_load_async_to_lds_b128 %0, %1, off"
                     :: "v"(ldsa), "v"(ga) : "memory");
    }

    for (int i = 0; i < NMS_N; ++i) {
        // --- removed-bit test (uniform indices -> readlane broadcast) ---
        int wi    = i >> 5;                           // removed-word index 0..127
        int owner = wi >> 2;                          // lane holding that word
        int comp  = wi & 3;                           // which uint4 component
        unsigned v;
        switch (comp) {
            case 0:  v = racc.x; break;
            case 1:  v = racc.y; break;
            case 2:  v = racc.z; break;
            default: v = racc.w; break;
        }
        unsigned word = __shfl(v, owner, 32);

        // Row i is the oldest of SWEEP_D outstanding async loads.
        asm volatile("s_wait_asynccnt 0x7" ::: "memory");

        if (((word >> (i & 31)) & 1u) == 0u) {        // box i survives -> suppress
            const uint4* row4 = (const uint4*)&buf[i & (SWEEP_D - 1)][0];
            uint4 t = row4[lane];                     // single ds_load_b128
            racc.x |= t.x;
            racc.y |= t.y;
            racc.z |= t.z;
            racc.w |= t.w;
        }

        // Refill the slot we just retired with row i+SWEEP_D.
        int nx = i + SWEEP_D;
        if (nx < NMS_N) {
            unsigned ldsa = lds0 + (unsigned)(nx & (SWEEP_D - 1)) * (NWORDS * 4u);
            unsigned long long ga =
                (unsigned long long)(uintptr_t)(mask + (size_t)nx * NWORDS) +
                (unsigned long long)(lane * 16);
            asm volatile("global_load_async_to_lds_b128 %0, %1, off"
                         :: "v"(ldsa), "v"(ga) : "memory");
        }
    }

    // keep word layout stays linear: word w lives at keepw[w], w = 4*lane+c.
    ((uint4*)keepw)[lane] = make_uint4(~racc.x, ~racc.y, ~racc.z, ~racc.w);
}

// ---------------------------------------------------------------------------
// Kernel 4: scatter back to original order.
// out[0 .. 4N)   = rois * keep   (kept_rois, row-major float4)
// out[4N .. 5N)  = keep as 0.0/1.0
// Each sorted slot s maps to a unique orig index (permutation) -> race-free.
// ---------------------------------------------------------------------------
__global__ __launch_bounds__(256) void nms_out(const float4* __restrict__ rois4,
                                               const unsigned* __restrict__ order,
                                               const unsigned* __restrict__ keepw,
                                               float* __restrict__ out)
{
    unsigned s    = blockIdx.x * 256u + threadIdx.x;  // sorted position
    unsigned orig = order[s];
    float k = (float)((keepw[s >> 5] >> (s & 31)) & 1u);
    float4 b = rois4[orig];
    float4* out_rois = (float4*)out;
    out_rois[orig] = make_float4(b.x * k, b.y * k, b.z * k, b.w * k);
    out[4u * NMS_N + orig] = k;
}

// ---------------------------------------------------------------------------
// Workspace layout (needs 65536 + 16384 + 512 + 2MB ~= 2.08 MB):
//   [0,       65536)   sorted boxes, float4[4096]
//   [65536,   81920)   order,        u32[4096]
//   [81920,   82432)   keep words,   u32[128]   (16B aligned)
//   [82432,   +2MB )   suppression mask, u32[4096*128]
// ---------------------------------------------------------------------------
extern "C" void kernel_launch(void* const* d_in, const int* in_sizes, int n_in,
                              void* d_out, int out_size, void* d_ws, size_t ws_size,
                              hipStream_t stream)
{
    const float* rois   = (const float*)d_in[0];      // [4096, 4] f32
    const float* scores = (const float*)d_in[1];      // [4096]   f32
    float* out = (float*)d_out;

    char* ws = (char*)d_ws;
    float4*   boxes = (float4*)ws;
    unsigned* order = (unsigned*)(ws + 65536);
    unsigned* keepw = (unsigned*)(ws + 65536 + 16384);
    unsigned* mask  = (unsigned*)(ws + 65536 + 16384 + 512);

    nms_sort <<<1, 1024, 0, stream>>>(scores, (const float4*)rois, order, boxes);
    nms_mask <<<(NMS_N * NWORDS) / 256, 256, 0, stream>>>(boxes, mask);
    nms_sweep<<<1, 32, 0, stream>>>(mask, keepw);
    nms_out  <<<NMS_N / 256, 256, 0, stream>>>((const float4*)rois, order, keepw, out);
}